// PhiAttention_5403068858742
// MI455X (gfx1250) — compile-verified
//
#include <hip/hip_runtime.h>

// ---------------- problem constants ----------------
constexpr int B_SZ   = 2;
constexpr int S_LEN  = 1024;
constexpr int T_TOK  = B_SZ * S_LEN;   // 2048
constexpr int HID_C  = 2560;
constexpr int H_NUM  = 32;
constexpr int D_HEAD = HID_C / H_NUM;  // 80
constexpr int ROT_C  = 32;             // int(0.4 * 80)
constexpr float THETA_C = 10000.0f;

typedef __attribute__((ext_vector_type(2))) float v2f;
typedef __attribute__((ext_vector_type(8))) float v8f;
typedef __attribute__((ext_vector_type(4))) int   v4i;

__device__ inline v8f wmma4(v2f a, v2f b, v8f c) {
  // D = A(16x4, f32) * B(4x16, f32) + C(16x16, f32)
  return __builtin_amdgcn_wmma_f32_16x16x4_f32(
      /*neg_a=*/false, a, /*neg_b=*/false, b,
      /*c_mod=*/(short)0, c, /*reuse_a=*/false, /*reuse_b=*/false);
}

// ---------------- CDNA5 async global->LDS staging ----------------
#if defined(__has_builtin)
#  if __has_builtin(__builtin_amdgcn_global_load_async_to_lds_b128)
#    define HAVE_ASYNC_LDS 1
#  endif
#  if __has_builtin(__builtin_amdgcn_s_wait_asynccnt)
#    define HAVE_WAIT_ASYNC_BUILTIN 1
#  endif
#endif

typedef __attribute__((address_space(1))) v4i gv4i;  // global int4
typedef __attribute__((address_space(3))) v4i lv4i;  // LDS int4

// One 16-byte element per call. With HAVE_ASYNC_LDS this lowers to
// global_load_async_to_lds_b128 (ASYNCcnt-tracked DMA into LDS).
__device__ inline void copy16_to_lds(const float* __restrict__ g, float* l) {
#ifdef HAVE_ASYNC_LDS
  gv4i* gp = (gv4i*)(uintptr_t)g;
  // low 32 bits of a generic LDS pointer are the LDS byte offset
  lv4i* lp = (lv4i*)(unsigned int)(uintptr_t)l;
  __builtin_amdgcn_global_load_async_to_lds_b128(gp, lp, 0, 0);
#else
  const float4 t = *(const float4*)g;
  l[0] = t.x; l[1] = t.y; l[2] = t.z; l[3] = t.w;
#endif
}

template <int N> __device__ inline void wait_async_le() {
#ifdef HAVE_ASYNC_LDS
#  ifdef HAVE_WAIT_ASYNC_BUILTIN
  __builtin_amdgcn_s_wait_asynccnt(N);
#  else
  asm volatile("s_wait_asynccnt %0" :: "i"(N) : "memory");
#  endif
#endif
}

// =====================================================================
// Tiled GEMM with fused bias:  C[M,N] = A[M,K] * W[K,N] + bias[N]
// 128 threads (4 waves), 64x64 tile, double-buffered async LDS staging.
// Each wave issues exactly 4 async b128 ops per stage, so
// s_wait_asynccnt<=4 retires the consumed stage while prefetch flies.
// =====================================================================
#define TM 64
#define TN 64
#define TK 16
#define APAD 4   // keeps 16B alignment for async b128 LDS writes
                 // (row strides 20 / 68 dwords; 20*i mod 64 distinct, i<16)

__global__ __launch_bounds__(128) void gemm_bias_wmma(
    const float* __restrict__ A, const float* __restrict__ W,
    const float* __restrict__ bias, float* __restrict__ C,
    int M, int N, int K)
{
  __shared__ float Als[2][TM][TK + APAD];
  __shared__ float Bls[2][TK][TN + APAD];

  const int tid  = threadIdx.x;
  const int wave = tid >> 5;
  const int lane = tid & 31;
  const int wm   = wave >> 1;            // 0..1  (wave row)
  const int wn   = wave & 1;             // 0..1  (wave col)
  const int r16  = lane & 15;
  const int khi  = (lane >> 4) << 1;     // 0 or 2 (A/B fragment K split)
  const int ml   = (lane >> 4) << 3;     // 0 or 8 (C fragment M split)

  const int m0 = blockIdx.y * TM;
  const int n0 = blockIdx.x * TN;

  v8f acc[2][2];
  acc[0][0] = {}; acc[0][1] = {}; acc[1][0] = {}; acc[1][1] = {};

  auto load_stage = [&](int bufi, int k0) {
    #pragma unroll
    for (int i = 0; i < 2; ++i) {            // A tile 64x16
      const int lin = tid + i * 128;         // 0..255 float4s
      const int row = lin >> 2;
      const int c4  = (lin & 3) << 2;
      copy16_to_lds(A + (size_t)(m0 + row) * K + k0 + c4, &Als[bufi][row][c4]);
    }
    #pragma unroll
    for (int i = 0; i < 2; ++i) {            // B tile 16x64
      const int lin = tid + i * 128;
      const int row = lin >> 4;
      const int c4  = (lin & 15) << 2;
      copy16_to_lds(W + (size_t)(k0 + row) * N + n0 + c4, &Bls[bufi][row][c4]);
    }
  };

  load_stage(0, 0);
  int bufi = 0;
  for (int k0 = 0; k0 < K; k0 += TK) {
    if (k0 + TK < K) {
      load_stage(bufi ^ 1, k0 + TK);   // prefetch next stage
      wait_async_le<4>();              // retire only the current stage
    } else {
      wait_async_le<0>();
    }
    __syncthreads();

    // Preload all 16 fragments (one LDS clause), then back-to-back WMMA.
    v2f af[4][2], bfr[4][2];
    #pragma unroll
    for (int kq = 0; kq < 4; ++kq) {
      const int kc = kq * 4 + khi;
      af[kq][0].x  = Als[bufi][wm * 32 + r16     ][kc];
      af[kq][0].y  = Als[bufi][wm * 32 + r16     ][kc + 1];
      af[kq][1].x  = Als[bufi][wm * 32 + 16 + r16][kc];
      af[kq][1].y  = Als[bufi][wm * 32 + 16 + r16][kc + 1];
      bfr[kq][0].x = Bls[bufi][kc    ][wn * 32 + r16];
      bfr[kq][0].y = Bls[bufi][kc + 1][wn * 32 + r16];
      bfr[kq][1].x = Bls[bufi][kc    ][wn * 32 + 16 + r16];
      bfr[kq][1].y = Bls[bufi][kc + 1][wn * 32 + 16 + r16];
    }
    #pragma unroll
    for (int kq = 0; kq < 4; ++kq) {
      acc[0][0] = wmma4(af[kq][0], bfr[kq][0], acc[0][0]);
      acc[0][1] = wmma4(af[kq][0], bfr[kq][1], acc[0][1]);
      acc[1][0] = wmma4(af[kq][1], bfr[kq][0], acc[1][0]);
      acc[1][1] = wmma4(af[kq][1], bfr[kq][1], acc[1][1]);
    }
    __syncthreads();
    bufi ^= 1;
  }

  #pragma unroll
  for (int i = 0; i < 2; ++i)
    #pragma unroll
    for (int j = 0; j < 2; ++j)
      #pragma unroll
      for (int e = 0; e < 8; ++e) {
        const int gm = m0 + wm * 32 + i * 16 + ml + e;
        const int gn = n0 + wn * 32 + j * 16 + r16;
        C[(size_t)gm * N + gn] = acc[i][j][e] + bias[gn];
      }
}

// =====================================================================
// RoPE in-place on q and k: rotate first ROT dims of each head.
// =====================================================================
__global__ void rope_kernel(float* __restrict__ q, float* __restrict__ k,
                            const int* __restrict__ positions)
{
  const int idx = blockIdx.x * blockDim.x + threadIdx.x;
  if (idx >= T_TOK * H_NUM) return;
  const int t = idx / H_NUM;
  const int h = idx % H_NUM;
  const float pos = (float)positions[t];
  const int base = t * HID_C + h * D_HEAD;
  #pragma unroll
  for (int i = 0; i < ROT_C / 2; ++i) {
    const float inv = __powf(THETA_C, -(2.0f * i) / (float)ROT_C);
    const float ang = pos * inv;
    const float c = __cosf(ang), s = __sinf(ang);
    const float q1 = q[base + i], q2 = q[base + ROT_C / 2 + i];
    q[base + i]             = q1 * c - q2 * s;
    q[base + ROT_C / 2 + i] = q2 * c + q1 * s;
    const float k1 = k[base + i], k2 = k[base + ROT_C / 2 + i];
    k[base + i]             = k1 * c - k2 * s;
    k[base + ROT_C / 2 + i] = k2 * c + k1 * s;
  }
}

// =====================================================================
// Flash attention, one wave per (b, h, 16-query tile).
// =====================================================================
__global__ __launch_bounds__(32) void attn_kernel(
    const float* __restrict__ q, const float* __restrict__ kk,
    const float* __restrict__ vbuf, float* __restrict__ o)
{
  __shared__ float Pls[16][17];

  const int lane = threadIdx.x;
  const int r16  = lane & 15;
  const int khi  = (lane >> 4) << 1;   // 0 / 2
  const int ml   = (lane >> 4) << 3;   // 0 / 8

  const int qt = blockIdx.x;           // query tile (16 rows)
  const int h  = blockIdx.y;
  const int b  = blockIdx.z;
  const int q0 = qt * 16;
  const float scale = 0.11180339887498949f;  // 80^-0.5

  // Q fragments (A layout), prescaled: 20 K-chunks of 4 over D=80
  v2f qa[20];
  {
    const int qtok = b * S_LEN + q0 + r16;
    const float* qrow = q + (size_t)qtok * HID_C + h * D_HEAD;
    #pragma unroll
    for (int c = 0; c < 20; ++c) {
      const float2 t = *(const float2*)(qrow + c * 4 + khi);
      qa[c].x = t.x * scale;
      qa[c].y = t.y * scale;
    }
  }

  v8f accO[5];
  #pragma unroll
  for (int t = 0; t < 5; ++t) accO[t] = {};
  float mrun[8], lrun[8];
  #pragma unroll
  for (int e = 0; e < 8; ++e) { mrun[e] = -3.0e38f; lrun[e] = 0.0f; }

  for (int j = 0; j <= qt; ++j) {
    // ---- S = (Q*scale) K^T for this 16x16 tile ----
    v8f s = {};
    const int ktok = b * S_LEN + j * 16 + r16;
    const float* krow = kk + (size_t)ktok * HID_C + h * D_HEAD;
    #pragma unroll
    for (int c = 0; c < 20; ++c) {
      const float2 t = *(const float2*)(krow + c * 4 + khi);
      v2f bf; bf.x = t.x; bf.y = t.y;
      s = wmma4(qa[c], bf, s);
    }

    // ---- causal mask on the diagonal tile ----
    if (j == qt) {
      #pragma unroll
      for (int e = 0; e < 8; ++e)
        if (r16 > ml + e) s[e] = -1.0e9f;
    }

    // ---- online softmax (row = ml+e, spread over 16 lanes) ----
    float p[8], corr[8];
    #pragma unroll
    for (int e = 0; e < 8; ++e) {
      float rm = s[e];
      rm = fmaxf(rm, __shfl_xor(rm, 1, 16));
      rm = fmaxf(rm, __shfl_xor(rm, 2, 16));
      rm = fmaxf(rm, __shfl_xor(rm, 4, 16));
      rm = fmaxf(rm, __shfl_xor(rm, 8, 16));
      const float mnew = fmaxf(mrun[e], rm);
      const float pe = __expf(s[e] - mnew);
      float rs = pe;
      rs += __shfl_xor(rs, 1, 16);
      rs += __shfl_xor(rs, 2, 16);
      rs += __shfl_xor(rs, 4, 16);
      rs += __shfl_xor(rs, 8, 16);
      const float cf = __expf(mrun[e] - mnew);
      lrun[e] = lrun[e] * cf + rs;
      mrun[e] = mnew;
      p[e] = pe;
      corr[e] = cf;
    }
    #pragma unroll
    for (int t = 0; t < 5; ++t)
      #pragma unroll
      for (int e = 0; e < 8; ++e)
        accO[t][e] *= corr[e];

    // ---- P: C-layout -> A-layout via LDS ----
    #pragma unroll
    for (int e = 0; e < 8; ++e)
      Pls[ml + e][r16] = p[e];
    __syncthreads();

    // ---- O += P * V  (B fragments streamed from global V) ----
    #pragma unroll
    for (int c = 0; c < 4; ++c) {
      v2f pa;
      pa.x = Pls[r16][c * 4 + khi];
      pa.y = Pls[r16][c * 4 + khi + 1];
      const float* vrow0 =
          vbuf + (size_t)(b * S_LEN + j * 16 + c * 4 + khi) * HID_C + h * D_HEAD;
      const float* vrow1 = vrow0 + HID_C;
      #pragma unroll
      for (int t = 0; t < 5; ++t) {
        v2f bf;
        bf.x = vrow0[t * 16 + r16];
        bf.y = vrow1[t * 16 + r16];
        accO[t] = wmma4(pa, bf, accO[t]);
      }
    }
    __syncthreads();
  }

  // ---- finalize: O / l ----
  const int otok = b * S_LEN + q0;
  #pragma unroll
  for (int t = 0; t < 5; ++t)
    #pragma unroll
    for (int e = 0; e < 8; ++e) {
      const int row = otok + ml + e;
      o[(size_t)row * HID_C + h * D_HEAD + t * 16 + r16] = accO[t][e] / lrun[e];
    }
}

// =====================================================================
// Launch: QKV GEMMs -> RoPE -> flash attention -> dense GEMM.
// Cache writes never affect the returned tensor, so they are skipped.
// =====================================================================
extern "C" void kernel_launch(void* const* d_in, const int* in_sizes, int n_in,
                              void* d_out, int out_size, void* d_ws, size_t ws_size,
                              hipStream_t stream)
{
  const float* h    = (const float*)d_in[0];
  const float* q_w  = (const float*)d_in[3];
  const float* q_b  = (const float*)d_in[4];
  const float* k_w  = (const float*)d_in[5];
  const float* k_b  = (const float*)d_in[6];
  const float* v_w  = (const float*)d_in[7];
  const float* v_b  = (const float*)d_in[8];
  const float* d_wm = (const float*)d_in[9];
  const float* d_bv = (const float*)d_in[10];
  const int* positions = (const int*)d_in[11];

  const size_t buf = (size_t)T_TOK * HID_C;
  float* qbuf = (float*)d_ws;
  float* kbuf = qbuf + buf;
  float* vbuf = kbuf + buf;
  float* obuf = qbuf;  // alias is race-free: each wave reads only its own
                       // q rows (once, up front) and writes only those rows.

  dim3 ggrid(HID_C / TN, T_TOK / TM);
  gemm_bias_wmma<<<ggrid, 128, 0, stream>>>(h, q_w, q_b, qbuf, T_TOK, HID_C, HID_C);
  gemm_bias_wmma<<<ggrid, 128, 0, stream>>>(h, k_w, k_b, kbuf, T_TOK, HID_C, HID_C);
  gemm_bias_wmma<<<ggrid, 128, 0, stream>>>(h, v_w, v_b, vbuf, T_TOK, HID_C, HID_C);

  rope_kernel<<<(T_TOK * H_NUM + 255) / 256, 256, 0, stream>>>(qbuf, kbuf, positions);

  attn_kernel<<<dim3(S_LEN / 16, H_NUM, B_SZ), 32, 0, stream>>>(qbuf, kbuf, vbuf, obuf);

  gemm_bias_wmma<<<ggrid, 128, 0, stream>>>(obuf, d_wm, d_bv, (float*)d_out,
                                            T_TOK, HID_C, HID_C);
}